// Loss_14714557956386
// MI455X (gfx1250) — compile-verified
//
#include <hip/hip_runtime.h>
#include <hip/hip_bf16.h>
#include <stdint.h>

// Problem constants (match reference).
#define N_WORDS   4096
#define N_SAMPLES 1024
#define NPAIRS    4095           // pairs per sample row
#define INV       (1.0f / 1024.0f)

typedef __attribute__((ext_vector_type(2))) float  v2f;
typedef __attribute__((ext_vector_type(8))) float  v8f;
typedef __attribute__((ext_vector_type(4))) unsigned int v4u;
typedef __attribute__((ext_vector_type(8))) int    v8i;
typedef __attribute__((ext_vector_type(4))) int    v4i;

// ---------------------------------------------------------------------------
// Kernel 1: one block per sample row.
//   - TDM (tensor_load_to_lds) DMAs the 4096-int sample row into LDS.
//   - Each thread gathers 16 (a,b) pairs from bigram & bigram_bias (random
//     4B gathers; 128MB of tables is L2-resident on MI455X's 192MB L2).
//   - Per-wave reduction via V_WMMA_F32_16X16X4_F32 (A = lane accumulators,
//     B = ones), then per-block partial to d_ws.
// ---------------------------------------------------------------------------
__global__ __launch_bounds__(256) void pair_gather_kernel(
    const float* __restrict__ bigram,
    const float* __restrict__ bias,
    const int*   __restrict__ samples,
    float*       __restrict__ partials) {
  __shared__ int   sm[N_WORDS];
  __shared__ float wred[8];

  const int s = blockIdx.x;
  const int t = threadIdx.x;

  // ---- Stage sample row into LDS -----------------------------------------
#if __has_builtin(__builtin_amdgcn_tensor_load_to_lds) && __has_builtin(__builtin_amdgcn_s_wait_tensorcnt)
  if (t < 32) {  // wave 0 issues the TDM descriptor once
    uint64_t ga  = (uint64_t)(uintptr_t)(samples + (size_t)s * N_WORDS);
    uint32_t lds = (uint32_t)(uintptr_t)(&sm[0]);  // low 32 bits of flat = LDS byte offset

    // D# group 0: count=1 | lds_addr | global_addr[56:0] | type=2
    v4u g0;
    g0.x = 1u;
    g0.y = lds;
    g0.z = (uint32_t)ga;
    g0.w = (uint32_t)((ga >> 32) & 0x1FFFFFFull) | 0x80000000u;

    // D# group 1: data_size=4B; tensor_dim0=4096; tensor_dim1=1;
    //             tile_dim0=4096 (1 row); tile_dim1/2 unused; stride0=4096.
    v8i g1;
    g1[0] = 0x00020000;                       // workgroup_mask=0, data_size=2 (4B)
    g1[1] = (int)((N_WORDS & 0xFFFF) << 16);  // tensor_dim0[15:0] in bits[31:16]
    g1[2] = (int)(((N_WORDS >> 16) & 0xFFFF) | (1 << 16)); // dim0 hi | tensor_dim1=1
    g1[3] = (int)(N_WORDS << 16);             // tensor_dim1 hi=0 | tile_dim0=4096
    g1[4] = 0;                                // tile_dim1=0, tile_dim2=0 (1D tile)
    g1[5] = N_WORDS;                          // tensor_dim0_stride lo
    g1[6] = 0;
    g1[7] = 0;

    v4i z4 = {0, 0, 0, 0};                    // groups 2/3 unused (<=2D tensor)
    v8i z8 = {0, 0, 0, 0, 0, 0, 0, 0};        // 6-arg toolchain extra group
    __builtin_amdgcn_tensor_load_to_lds(g0, g1, z4, z4, z8, 0);
    __builtin_amdgcn_s_wait_tensorcnt(0);
  }
  __syncthreads();
#else
  for (int i = t; i < N_WORDS; i += 256)
    sm[i] = samples[(size_t)s * N_WORDS + i];
  __syncthreads();
#endif

  // ---- Random gathers (the real work) ------------------------------------
  float acc = 0.0f;
#pragma unroll
  for (int it = 0; it < 16; ++it) {
    int j = it * 256 + t;
    if (j < NPAIRS) {
      unsigned a = (unsigned)sm[j];
      unsigned b = (unsigned)sm[j + 1];
      unsigned idx = (a << 12) | b;           // a*4096 + b (b < 4096)
      acc += bigram[idx] + bias[idx];
    }
  }

  // ---- Wave reduction via f32 WMMA ---------------------------------------
  float wsum;
#if __has_builtin(__builtin_amdgcn_wmma_f32_16x16x4_f32)
  {
    // A (16x4 f32): lane m V0 -> A[m,0] (lanes 0-15), A[m,2] (lanes 16-31);
    // V1 = 0 -> K=1,3 columns zero. B = ones => D[m,n] = acc(m) + acc(m+16).
    v2f A; A.x = acc;  A.y = 0.0f;
    v2f B; B.x = 1.0f; B.y = 1.0f;
    v8f C = {};
    C = __builtin_amdgcn_wmma_f32_16x16x4_f32(false, A, false, B,
                                              (short)0, C, false, false);
    float h = C[0] + C[1] + C[2] + C[3] + C[4] + C[5] + C[6] + C[7];
    // lanes 0-15 hold sum over M=0..7, lanes 16-31 hold sum over M=8..15
    wsum = h + __shfl_xor(h, 16, 32);
  }
#else
  wsum = acc;
  for (int o = 16; o > 0; o >>= 1) wsum += __shfl_xor(wsum, o, 32);
#endif

  const int lane = t & 31, wid = t >> 5;
  if (lane == 0) wred[wid] = wsum;
  __syncthreads();
  if (t == 0) {
    float p = 0.0f;
#pragma unroll
    for (int w = 0; w < 8; ++w) p += wred[w];
    partials[blockIdx.x] = p;   // un-scaled sum over sampled pairs, both tables
  }
}

// ---------------------------------------------------------------------------
// Kernel 2: single-block deterministic finalize.
//   loss = INV*(sum(partials) + start/end sample terms)
//        - gold chain (both tables) - start[0] - end[4095]
// ---------------------------------------------------------------------------
__global__ __launch_bounds__(256) void finalize_kernel(
    const float* __restrict__ bigram,
    const float* __restrict__ start,
    const float* __restrict__ endv,
    const float* __restrict__ bias,
    const int*   __restrict__ samples,
    const float* __restrict__ partials,
    float*       __restrict__ out,
    int outn) {
  __shared__ float red[256];
  const int t = threadIdx.x;

  float ps = 0.0f;
  for (int i = t; i < N_SAMPLES; i += 256) ps += partials[i];

  float gold = 0.0f;
  for (int i = t; i < NPAIRS; i += 256) {
    unsigned idx = ((unsigned)i << 12) + (unsigned)i + 1u;  // tbl[i, i+1]
    gold += bigram[idx] + bias[idx];
  }

  float se = 0.0f;
  for (int s = t; s < N_SAMPLES; s += 256) {
    se += start[samples[(size_t)s * N_WORDS]];
    se += endv[samples[(size_t)s * N_WORDS + NPAIRS]];
  }

  red[t] = INV * (ps + se) - gold;
  __syncthreads();
  for (int h = 128; h > 0; h >>= 1) {
    if (t < h) red[t] += red[t + h];
    __syncthreads();
  }
  if (t == 0) {
    out[0] = red[0] - start[0] - endv[N_WORDS - 1];
    if (outn > 1) out[1] = 0.0f;
  }
}

// ---------------------------------------------------------------------------
extern "C" void kernel_launch(void* const* d_in, const int* in_sizes, int n_in,
                              void* d_out, int out_size, void* d_ws, size_t ws_size,
                              hipStream_t stream) {
  const float* bigram  = (const float*)d_in[0];
  const float* start   = (const float*)d_in[1];
  const float* endv    = (const float*)d_in[2];
  const float* bias    = (const float*)d_in[3];
  const int*   samples = (const int*)d_in[4];
  float* partials = (float*)d_ws;   // N_SAMPLES floats = 4 KB
  float* out = (float*)d_out;

  pair_gather_kernel<<<N_SAMPLES, 256, 0, stream>>>(bigram, bias, samples, partials);
  finalize_kernel<<<1, 256, 0, stream>>>(bigram, start, endv, bias, samples,
                                         partials, out, out_size);
}